// GCN_21569325760838
// MI455X (gfx1250) — compile-verified
//
#include <hip/hip_runtime.h>

// GCN, 2 layers: N=100000 nodes, E=1.6M edges, 128 -> 128(ReLU) -> 64.
// fp32 throughout (matches reference numerics); dense GEMMs use
// V_WMMA_F32_16X16X4_F32 with a software-pipelined LDS B-feed; aggregation
// uses native f32 global atomics (accumulators are L2-resident: 51 MB < 192 MB).

#define NF 128   // input feats  (= K of GEMM1, N of GEMM1 output)
#define NH 128   // hidden feats (= K of GEMM2)
#define NO 64    // output feats

typedef float v2f __attribute__((ext_vector_type(2)));
typedef float v8f __attribute__((ext_vector_type(8)));

__device__ __forceinline__ void atomAddF(float* p, float v) {
  unsafeAtomicAdd(p, v);   // -> global_atomic_add_f32
}

// ---------------- degree / symmetric norm ----------------
__global__ void k_deg_init(float* __restrict__ deg, int n) {
  int i = blockIdx.x * blockDim.x + threadIdx.x;
  if (i < n) deg[i] = 1.0f;                    // self-loop contribution
}
__global__ void k_deg_edges(const int* __restrict__ dst,
                            float* __restrict__ deg, int e) {
  int i = blockIdx.x * blockDim.x + threadIdx.x;
  if (i < e) atomAddF(&deg[dst[i]], 1.0f);
}
__global__ void k_dinv(float* __restrict__ deg, int n) {
  int i = blockIdx.x * blockDim.x + threadIdx.x;
  if (i < n) deg[i] = rsqrtf(deg[i]);          // deg >= 1 always (self loop)
}

// ---------------- dense GEMM: Y[nrows x NOUT] = X[nrows x KDIM] @ W ----------
// One wave owns 16 rows x NOUT cols; block = 8 waves = 128 rows.
// W is staged transposed in LDS with a +4-float row pad: addr = col*(KDIM+4)+k,
// (KDIM+4) mod 64 == 4, so the 16 cols x 4 k-words touched by one b64 B-load
// land on 64 distinct banks (conflict-free).
// The k-loop is software-pipelined: next step's A (global b64) and all NT
// B fragments (ds b64) are fetched into a second register set before the
// current step's WMMAs issue, so LDS/HBM latency overlaps matrix math.
template<int KDIM, int NOUT>
__global__ __launch_bounds__(256)
void k_gemm_wmma(const float* __restrict__ X, const float* __restrict__ W,
                 float* __restrict__ Y, int nrows) {
  constexpr int LSTR = KDIM + 4;
  constexpr int NT   = NOUT / 16;
  extern __shared__ float lw[];                // [NOUT][LSTR], W transposed

  const int tid = threadIdx.x;
  // Cooperative transposed load: W[k][n] -> lw[n][k]. Coalesced global reads.
  for (int i = tid; i < KDIM * NOUT; i += 256) {
    int k = i / NOUT, n = i - k * NOUT;
    lw[n * LSTR + k] = W[i];
  }
  __syncthreads();

  const int wave = tid >> 5;
  const int lane = tid & 31;
  const int half = lane >> 4;                  // lane 16..31 -> K+2 slice
  const int lr   = lane & 15;
  const int koff = 2 * half;
  const int rowBase = blockIdx.x * 128 + wave * 16;

  v8f acc[NT];
#pragma unroll
  for (int t = 0; t < NT; ++t)
#pragma unroll
    for (int j = 0; j < 8; ++j) acc[t][j] = 0.0f;

  int arow = rowBase + lr;
  if (arow >= nrows) arow = nrows - 1;         // clamp: keeps EXEC all-1s
  const float* xrow = X + (size_t)arow * KDIM;
  const float* lwp  = lw + lr * LSTR;          // per-lane LDS base; t*16*LSTR
                                               // folds into ds_load offsets

  // Prologue: fragments for k0 = 0.
  v2f a_cur = *(const v2f*)(xrow + koff);
  v2f b_cur[NT];
#pragma unroll
  for (int t = 0; t < NT; ++t)
    b_cur[t] = *(const v2f*)(lwp + t * 16 * LSTR + koff);

#pragma unroll 2
  for (int k0 = 0; k0 < KDIM; k0 += 4) {
    // Prefetch next step's fragments (wraps to k=0 on last iter; dead loads).
    const int kn = ((k0 + 4) & (KDIM - 1)) + koff;
    v2f a_nxt = *(const v2f*)(xrow + kn);
    v2f b_nxt[NT];
#pragma unroll
    for (int t = 0; t < NT; ++t)
      b_nxt[t] = *(const v2f*)(lwp + t * 16 * LSTR + kn);

    // Current step's WMMAs consume the previously-fetched register set.
#pragma unroll
    for (int t = 0; t < NT; ++t)
      acc[t] = __builtin_amdgcn_wmma_f32_16x16x4_f32(
          false, a_cur, false, b_cur[t], (short)0, acc[t], false, false);

    a_cur = a_nxt;
#pragma unroll
    for (int t = 0; t < NT; ++t) b_cur[t] = b_nxt[t];
  }

  // D layout: VGPR v -> row rowBase + v + 8*half, col = t*16 + lr
#pragma unroll
  for (int t = 0; t < NT; ++t) {
    const int col = t * 16 + lr;
#pragma unroll
    for (int v = 0; v < 8; ++v) {
      const int row = rowBase + v + 8 * half;
      if (row < nrows) Y[(size_t)row * NOUT + col] = acc[t][v];
    }
  }
}

// ---------------- aggregation: out = D^-1/2 (A+I) D^-1/2 h ------------------
// Self-loop term initializes the accumulator: out[i] = h[i] * dinv[i]^2.
template<int F>
__global__ __launch_bounds__(256)
void k_agg_self(const float4* __restrict__ h4, const float* __restrict__ dinv,
                float4* __restrict__ out4, int n) {
  const int i = blockIdx.x * blockDim.x + threadIdx.x;
  const int total = n * (F / 4);
  if (i >= total) return;
  const int node = i / (F / 4);
  float w = dinv[node]; w *= w;
  float4 v = h4[i];
  out4[i] = make_float4(v.x * w, v.y * w, v.z * w, v.w * w);
}

// One edge per F/4-lane slice: full 4*F-byte source row in one coalesced shot,
// then 4 native f32 atomic adds per lane into the (L2-resident) accumulator.
template<int F>
__global__ __launch_bounds__(256)
void k_agg_edges(const int* __restrict__ src, const int* __restrict__ dst,
                 const float* __restrict__ dinv, const float* __restrict__ h,
                 float* __restrict__ out, int nedges) {
  constexpr int L = F / 4;                     // lanes per edge
  const int t = blockIdx.x * 256 + threadIdx.x;
  const int e = t / L;
  const int c = t - e * L;
  if (e >= nedges) return;
  const int s = src[e], d = dst[e];
  const float w = dinv[s] * dinv[d];
  float4 v = ((const float4*)(h + (size_t)s * F))[c];
  float* o = out + (size_t)d * F + c * 4;
  atomAddF(o + 0, v.x * w);
  atomAddF(o + 1, v.y * w);
  atomAddF(o + 2, v.z * w);
  atomAddF(o + 3, v.w * w);
}

template<int F, bool RELU>
__global__ __launch_bounds__(256)
void k_finalize(float4* __restrict__ out4, const float4* __restrict__ bias4,
                int n) {
  const int i = blockIdx.x * blockDim.x + threadIdx.x;
  const int total = n * (F / 4);
  if (i >= total) return;
  float4 b = bias4[i & (F / 4 - 1)];
  float4 v = out4[i];
  v.x += b.x; v.y += b.y; v.z += b.z; v.w += b.w;
  if (RELU) {
    v.x = fmaxf(v.x, 0.0f); v.y = fmaxf(v.y, 0.0f);
    v.z = fmaxf(v.z, 0.0f); v.w = fmaxf(v.w, 0.0f);
  }
  out4[i] = v;
}

static inline int cdiv(long long a, long long b) { return (int)((a + b - 1) / b); }

extern "C" void kernel_launch(void* const* d_in, const int* in_sizes, int n_in,
                              void* d_out, int out_size, void* d_ws, size_t ws_size,
                              hipStream_t stream) {
  const int*   edge = (const int*)d_in[0];     // [2, E] row-major
  const float* x    = (const float*)d_in[1];   // [N, 128]
  const float* W1   = (const float*)d_in[2];   // [128, 128]
  const float* b1   = (const float*)d_in[3];   // [128]
  const float* W2   = (const float*)d_in[4];   // [128, 64]
  const float* b2   = (const float*)d_in[5];   // [64]
  const int E = in_sizes[0] / 2;
  const int N = in_sizes[1] / NF;
  const int* src = edge;
  const int* dst = edge + E;

  // Workspace: dinv[N] | bufA[N*128] | bufB[N*128]   (~103 MB)
  float* dinv = (float*)d_ws;
  float* bufA = dinv + N;                      // N*4 bytes keeps 16B alignment
  float* bufB = bufA + (size_t)N * NF;

  // --- normalization: deg(A+I) -> rsqrt, in place ---
  k_deg_init <<<cdiv(N, 256), 256, 0, stream>>>(dinv, N);
  k_deg_edges<<<cdiv(E, 256), 256, 0, stream>>>(dst, dinv, E);
  k_dinv     <<<cdiv(N, 256), 256, 0, stream>>>(dinv, N);

  // --- layer 1: hx = x @ W1 ; h1 = relu(agg(hx) + b1) ---
  k_gemm_wmma<NF, NH><<<cdiv(N, 128), 256, (NH * (NF + 4)) * sizeof(float), stream>>>(
      x, W1, bufA, N);
  k_agg_self<NH><<<cdiv((long long)N * (NH / 4), 256), 256, 0, stream>>>(
      (const float4*)bufA, dinv, (float4*)bufB, N);
  k_agg_edges<NH><<<cdiv((long long)E * (NH / 4), 256), 256, 0, stream>>>(
      src, dst, dinv, bufA, bufB, E);
  k_finalize<NH, true><<<cdiv((long long)N * (NH / 4), 256), 256, 0, stream>>>(
      (float4*)bufB, (const float4*)b1, N);

  // --- layer 2: h2 = h1 @ W2 ; out = agg(h2) + b2 ---
  k_gemm_wmma<NH, NO><<<cdiv(N, 128), 256, (NO * (NH + 4)) * sizeof(float), stream>>>(
      bufB, W2, bufA, N);
  k_agg_self<NO><<<cdiv((long long)N * (NO / 4), 256), 256, 0, stream>>>(
      (const float4*)bufA, dinv, (float4*)d_out, N);
  k_agg_edges<NO><<<cdiv((long long)E * (NO / 4), 256), 256, 0, stream>>>(
      src, dst, dinv, bufA, (float*)d_out, E);
  k_finalize<NO, false><<<cdiv((long long)N * (NO / 4), 256), 256, 0, stream>>>(
      (float4*)d_out, (const float4*)b2, N);

  (void)n_in; (void)out_size; (void)ws_size;
}